// COPNLL_45715631899485
// MI455X (gfx1250) — compile-verified
//
#include <hip/hip_runtime.h>
#include <math.h>
#include <stdint.h>

#define NN 4096
#define NB 64
#define NBLK (NN / NB)

typedef __attribute__((ext_vector_type(2))) float v2f;
typedef __attribute__((ext_vector_type(8))) float v8f;

#if __has_builtin(__builtin_amdgcn_tensor_load_to_lds)
#define HAVE_TDM 1
#if __has_include(<hip/amd_detail/amd_gfx1250_TDM.h>)
#define TDM_6ARG 1
#endif
typedef unsigned int v4u __attribute__((ext_vector_type(4)));
typedef int v4i __attribute__((ext_vector_type(4)));
typedef int v8i __attribute__((ext_vector_type(8)));

// TDM load of a 64x64 f32 tile (row stride NN elements) into LDS at byte
// offset lds_off, padded 1 dword per 64 dwords -> matches [64][65] layout.
__device__ __forceinline__ void tdm_load_tile(const float* gsrc, unsigned lds_off) {
    unsigned long long ga = (unsigned long long)(uintptr_t)gsrc;
    v4u g0;
    g0[0] = 1u;                                  // count=1, user descriptor
    g0[1] = lds_off;                             // lds_addr (bytes)
    g0[2] = (unsigned)(ga & 0xffffffffu);        // global_addr[31:0]
    g0[3] = (unsigned)((ga >> 32) & 0x1ffffffu)  // global_addr[56:32]
          | (2u << 30);                          // type=2 ("image")
    v8i g1;
    g1[0] = (int)((2u << 16)     // data_size = 4B
                | (1u << 20)     // pad_enable
                | (5u << 22)     // pad_interval: 64 dwords
                | (0u << 25));   // pad_amount: 1 dword
    g1[1] = (int)(64u << 16);    // tensor_dim0 = 64 (bits 79:48, low half)
    g1[2] = (int)(64u << 16);    // tensor_dim1 = 64 (bits 111:80, low half)
    g1[3] = (int)(64u << 16);    // tile_dim0   = 64 (bits 127:112)
    g1[4] = 64;                  // tile_dim1   = 64 (bits 143:128)
    g1[5] = NN;                  // tensor_dim0_stride = 4096 (bits 207:160)
    g1[6] = 0;
    g1[7] = 0;
    v4i z = {0, 0, 0, 0};
#ifdef TDM_6ARG
    v8i z8 = {0, 0, 0, 0, 0, 0, 0, 0};
    __builtin_amdgcn_tensor_load_to_lds(g0, g1, z, z, z8, 0);
#else
    __builtin_amdgcn_tensor_load_to_lds(g0, g1, z, z, 0);
#endif
}
#endif

// scal[0]=sig2, [1]=sum_log_pdf, [2]=mtm, [3]=logdet, [4]=mVinvm
__global__ void k_init(const float* sig2e, const float* sig2bs, float* scal) {
    scal[0] = sig2e[0] + sig2bs[0] + sig2bs[1];
    scal[1] = 0.f; scal[2] = 0.f; scal[3] = 0.f; scal[4] = 0.f;
}

__device__ __forceinline__ float block_reduce(float v, float* buf) {
    int tid = threadIdx.x;
    buf[tid] = v;
    __syncthreads();
    for (int s = blockDim.x >> 1; s > 0; s >>= 1) {
        if (tid < s) buf[tid] += buf[tid + s];
        __syncthreads();
    }
    return buf[0];
}

// ------------------------------------------------- m, sum_log_pdf, m^T m
__global__ void k_resid(const float* yt, const float* yp, float* m, float* w,
                        float* scal) {
    __shared__ float r1[256], r2[256];
    int i = blockIdx.x * blockDim.x + threadIdx.x;
    float sig2 = scal[0];
    float r = yt[i] - yp[i];
    float u = 0.5f * (1.f + erff(r * rsqrtf(2.f * sig2)));
    u = fminf(fmaxf(u, 1e-5f), 1.f - 1e-5f);
    float mm = erfinvf(2.f * u - 1.f) * 1.41421356237309515f;
    m[i] = mm;
    w[i] = mm;
    float lp = -0.5f * logf(2.f * 3.14159265358979323846f * sig2)
             - r * r / (2.f * sig2);
    float slp = block_reduce(lp, r1);
    __syncthreads();
    float smt = block_reduce(mm * mm, r2);
    if (threadIdx.x == 0) {
        atomicAdd(&scal[1], slp);
        atomicAdd(&scal[2], smt);
    }
}

// ---------------------------------------------------------------- build V
__global__ void k_buildV(float* V, const int* z0, const int* z1,
                         const float* sig2e_p, const float* sig2bs_p,
                         const float* scal) {
    int i = blockIdx.y;
    int j0 = (blockIdx.x * blockDim.x + threadIdx.x) * 4;
    float inv = 1.f / scal[0];
    float se = sig2e_p[0] * inv;
    float s0 = sig2bs_p[0] * inv;
    float s1 = sig2bs_p[1] * inv;
    int a0 = z0[i], a1 = z1[i];
    float t[4];
#pragma unroll
    for (int q = 0; q < 4; ++q) {
        int j = j0 + q;
        float v = 0.f;
        if (a0 == z0[j]) v += s0;
        if (a1 == z1[j]) v += s1;
        if (i == j) v += se;
        t[q] = v;
    }
    float4 out; out.x = t[0]; out.y = t[1]; out.z = t[2]; out.w = t[3];
    *(float4*)(V + (size_t)i * NN + j0) = out;
}

// ---------------- fused step k: factor 64x64 diag (redundant per block,
// block 0 stores + logdet) then TRSM one panel block row per block.
// Right-looking forms split 4 ways per row to cut the serial FMA chain.
__global__ void __launch_bounds__(256) k_chol_panel(float* V, int k, int R,
                                                    float* scal) {
    __shared__ float a[NB][NB + 1];
    __shared__ float B[NB][NB + 1];
    int tid = threadIdx.x;
    int r = tid & 63, part = tid >> 6;       // 4 sub-threads per row
    for (int e = tid; e < NB * NB; e += 256) {
        int rr = e >> 6, c = e & 63;
        a[rr][c] = V[(size_t)(k * NB + rr) * NN + k * NB + c];
    }
    __syncthreads();
    for (int c = 0; c < NB; ++c) {
        if (tid == c) a[c][c] = sqrtf(a[c][c]);
        __syncthreads();
        if (part == 0 && r > c) a[r][c] /= a[c][c];
        __syncthreads();
        if (r > c) {
            float lc = a[r][c];
            for (int j = c + 1 + part; j < NB; j += 4) a[r][j] -= lc * a[j][c];
        }
        __syncthreads();
    }
    if (blockIdx.x == 0) {
        if (tid < NB) atomicAdd(&scal[3], 2.f * logf(a[tid][tid]));
        for (int e = tid; e < NB * NB; e += 256) {
            int rr = e >> 6, c = e & 63;
            V[(size_t)(k * NB + rr) * NN + k * NB + c] = a[rr][c];
        }
    }
    if (R == 0) return;
    int ib = k + 1 + blockIdx.x;
    for (int e = tid; e < NB * NB; e += 256) {
        int rr = e >> 6, c = e & 63;
        B[rr][c] = V[(size_t)(ib * NB + rr) * NN + k * NB + c];
    }
    __syncthreads();
    // Solve X * Lkk^T = B, right-looking over columns.
    for (int c = 0; c < NB; ++c) {
        if (part == 0) B[r][c] /= a[c][c];
        __syncthreads();
        float xc = B[r][c];
        for (int j = c + 1 + part; j < NB; j += 4) B[r][j] -= xc * a[j][c];
        __syncthreads();
    }
    for (int e = tid; e < NB * NB; e += 256) {
        int rr = e >> 6, c = e & 63;
        V[(size_t)(ib * NB + rr) * NN + k * NB + c] = B[rr][c];
    }
}

// ------------------- trailing update (WMMA f32): A[i,j] -= P_i * P_j^T
__global__ void __launch_bounds__(128) k_chol_update(float* V, int k) {
    int bi = k + 1 + blockIdx.x;
    int bj = k + 1 + blockIdx.y;
    if (bj > bi) return;   // uniform per block: lower triangle only
    __shared__ float P[2 * NB * (NB + 1)];
    float* Pi = P;
    float* Pj = P + NB * (NB + 1);
    int tid = threadIdx.x;
#ifdef HAVE_TDM
    if (tid < 32) {   // wave 0 issues both TDM tile loads, waits TENSORcnt
        tdm_load_tile(V + (size_t)(bi * NB) * NN + k * NB, 0u);
        tdm_load_tile(V + (size_t)(bj * NB) * NN + k * NB,
                      (unsigned)(NB * (NB + 1) * 4));
        __builtin_amdgcn_s_wait_tensorcnt(0);
    }
    // The TDM engine wrote LDS behind the compiler's back; escape the
    // pointer through an asm with a memory clobber so the LDS reads below
    // cannot be folded to undef (they were in the previous build).
    {
        float* p = P;
        asm volatile("" : "+v"(p) :: "memory");
    }
#else
    for (int e = tid; e < NB * NB; e += 128) {
        int r = e >> 6, c = e & 63;
        Pi[r * (NB + 1) + c] = V[(size_t)(bi * NB + r) * NN + k * NB + c];
        Pj[r * (NB + 1) + c] = V[(size_t)(bj * NB + r) * NN + k * NB + c];
    }
#endif
    __syncthreads();
    int wv = tid >> 5, lane = tid & 31;
    int qm = (wv >> 1) * 32;            // wave's 32x32 quadrant
    int qn = (wv & 1) * 32;
    int lm   = lane & 15;               // M (A) / N (B,C) within 16x16 tile
    int kb   = (lane >> 4) << 1;        // K base 0 / 2 (ISA 16x4 f32 layout)
    int rsel = (lane >> 4) << 3;        // C/D row base 0 / 8
    v8f acc[2][2] = {};
    for (int ks = 0; ks < NB; ks += 4) {
        v2f af[2], bf[2];
#pragma unroll
        for (int t = 0; t < 2; ++t) {
            int rm = qm + t * 16 + lm;  // A: P_i rows
            af[t].x = Pi[rm * (NB + 1) + ks + kb];
            af[t].y = Pi[rm * (NB + 1) + ks + kb + 1];
            int cn = qn + t * 16 + lm;  // B = P_j^T: B[k][n] = P_j[n][k]
            bf[t].x = Pj[cn * (NB + 1) + ks + kb];
            bf[t].y = Pj[cn * (NB + 1) + ks + kb + 1];
        }
#pragma unroll
        for (int tm = 0; tm < 2; ++tm)
#pragma unroll
            for (int tn = 0; tn < 2; ++tn)
                acc[tm][tn] = __builtin_amdgcn_wmma_f32_16x16x4_f32(
                    false, af[tm], false, bf[tn], (short)0, acc[tm][tn],
                    false, false);
    }
#pragma unroll
    for (int tm = 0; tm < 2; ++tm)
#pragma unroll
        for (int tn = 0; tn < 2; ++tn) {
            size_t col = (size_t)bj * NB + qn + tn * 16 + lm;
            int row0 = bi * NB + qm + tm * 16 + rsel;
#pragma unroll
            for (int v = 0; v < 8; ++v) {
                size_t idx = (size_t)(row0 + v) * NN + col;
                V[idx] -= acc[tm][tn][v];
            }
        }
}

// ---------------- invert all 64 diagonal blocks in parallel (one block per
// diag tile, one thread per column of inv(Lkk); column recurrences are
// thread-private so no barriers in the substitution loop).
__global__ void __launch_bounds__(NB) k_linv(const float* V, float* linv) {
    __shared__ float L[NB][NB + 1];
    __shared__ float X[NB][NB + 1];
    int kb = blockIdx.x, j = threadIdx.x;
    for (int e = j; e < NB * NB; e += NB) {
        int rr = e >> 6, c = e & 63;
        L[rr][c] = V[(size_t)(kb * NB + rr) * NN + kb * NB + c];
    }
    __syncthreads();
    for (int i = 0; i < NB; ++i) {
        float s = (i == j) ? 1.f : 0.f;
        for (int d = j; d < i; ++d) s -= L[i][d] * X[d][j];
        X[i][j] = (i >= j) ? s / L[i][i] : 0.f;
    }
    __syncthreads();
    for (int e = j; e < NB * NB; e += NB) {
        int rr = e >> 6, c = e & 63;
        linv[(size_t)kb * NB * NB + rr * NB + c] = X[rr][c];
    }
}

// ---------------- whole forward solve L y = w in ONE launch: per block-step,
// y_k = inv(Lkk) * w_k (parallel mat-vec, no scalar recurrence), then stream
// the panel update. V is L2-resident (64 MB < 192 MB L2).
__global__ void __launch_bounds__(1024) k_fwd_solve(const float* V,
                                                    const float* linv,
                                                    float* w, float* y) {
    __shared__ float yk[NB];
    int t = threadIdx.x;
    for (int kb = 0; kb < NBLK; ++kb) {
        if (t < NB) {
            const float* Li = linv + (size_t)kb * NB * NB + t * NB;
            float s = 0.f;
#pragma unroll
            for (int c = 0; c < NB; ++c) s += Li[c] * w[kb * NB + c];
            yk[t] = s;
            y[kb * NB + t] = s;
        }
        __syncthreads();
        for (int i = (kb + 1) * NB + t; i < NN; i += 1024) {
            const float* row = V + (size_t)i * NN + kb * NB;
            float s = 0.f;
#pragma unroll
            for (int c = 0; c < NB; ++c) s += row[c] * yk[c];
            w[i] -= s;
        }
        __syncthreads();
    }
}

// ---------------- whole backward solve L^T x = y in ONE launch (bottom-up,
// right-looking): x_k = inv(Lkk)^T * y_k, then y[0..kb*NB) -= L[kb-rows]^T x_k
// with coalesced row-major panel reads.
__global__ void __launch_bounds__(1024) k_bwd_solve(const float* V,
                                                    const float* linv,
                                                    float* y, float* x) {
    __shared__ float xk[NB];
    int t = threadIdx.x;
    for (int kb = NBLK - 1; kb >= 0; --kb) {
        if (t < NB) {
            float s = 0.f;
            for (int c = 0; c < NB; ++c)
                s += linv[(size_t)kb * NB * NB + c * NB + t] * y[kb * NB + c];
            xk[t] = s;
            x[kb * NB + t] = s;
        }
        __syncthreads();
        for (int c = t; c < kb * NB; c += 1024) {
            float s = 0.f;
#pragma unroll
            for (int i = 0; i < NB; ++i)
                s += V[(size_t)(kb * NB + i) * NN + c] * xk[i];
            y[c] -= s;
        }
        __syncthreads();
    }
}

// --------------------------------------------------------- dot and final
__global__ void k_dot(const float* m, const float* x, float* scal) {
    __shared__ float red[256];
    int tid = threadIdx.x;
    float s = 0.f;
    for (int i = blockIdx.x * 256 + tid; i < NN; i += gridDim.x * 256)
        s += m[i] * x[i];
    float tot = block_reduce(s, red);
    if (tid == 0) atomicAdd(&scal[4], tot);
}

__global__ void k_final(const float* scal, float* out) {
    // 0.5*logdet + 0.5*mVinvm - 0.5*mtm + 0.5*sum_log_pdf
    out[0] = 0.5f * (scal[3] + scal[4] - scal[2] + scal[1]);
}

extern "C" void kernel_launch(void* const* d_in, const int* in_sizes, int n_in,
                              void* d_out, int out_size, void* d_ws, size_t ws_size,
                              hipStream_t stream) {
    const float* y_true = (const float*)d_in[0];
    const float* y_pred = (const float*)d_in[1];
    const float* sig2e  = (const float*)d_in[2];
    const float* sig2bs = (const float*)d_in[3];
    const int*   z0     = (const int*)d_in[4];
    const int*   z1     = (const int*)d_in[5];
    float* out = (float*)d_out;

    float* V    = (float*)d_ws;                 // NN*NN f32 (64 MB)
    float* m    = V + (size_t)NN * NN;          // NN
    float* w    = m + NN;                       // NN  forward rhs (init = m)
    float* y    = w + NN;                       // NN  forward solution
    float* x    = y + NN;                       // NN  backward solution
    float* linv = x + NN;                       // 64 x 64x64 inv diag (1 MB)
    float* scal = linv + (size_t)NBLK * NB * NB;

    k_init<<<1, 1, 0, stream>>>(sig2e, sig2bs, scal);
    k_resid<<<NN / 256, 256, 0, stream>>>(y_true, y_pred, m, w, scal);
    k_buildV<<<dim3(NN / 1024, NN), 256, 0, stream>>>(V, z0, z1, sig2e, sig2bs, scal);

    for (int k = 0; k < NBLK; ++k) {
        int R = NBLK - 1 - k;
        k_chol_panel<<<(R > 0 ? R : 1), 256, 0, stream>>>(V, k, R, scal);
        if (R > 0)
            k_chol_update<<<dim3(R, R), 128, 0, stream>>>(V, k);
    }
    k_linv<<<NBLK, NB, 0, stream>>>(V, linv);
    k_fwd_solve<<<1, 1024, 0, stream>>>(V, linv, w, y);
    k_bwd_solve<<<1, 1024, 0, stream>>>(V, linv, y, x);
    k_dot<<<32, 256, 0, stream>>>(m, x, scal);
    k_final<<<1, 1, 0, stream>>>(scal, out);
}